// Net2_84954453115107
// MI455X (gfx1250) — compile-verified
//
#include <hip/hip_runtime.h>
#include <hip/hip_bf16.h>
#include <math.h>

// ---------------------------------------------------------------------------
// MI455X (gfx1250) implementation.
//
// Roofline: edge gather/scatter dominates (~2.3 GB @ 23.3 TB/s ~ 100us floor);
// the scatter-sum buffers (30-40MB) live in the 192MB L2, so f32 atomics are
// cheap. Dense GEMMs are only a few GMAC -> full-precision
// V_WMMA_F32_16X16X4_F32 (wave32, 16x16 tile, K-step 4); weights are
// pre-packed into zero-padded WMMA-fragment order so the GEMM inner loop is
// branch-free: 2x global_load_b64 + 1x v_wmma per K-step.
// ---------------------------------------------------------------------------

typedef __attribute__((ext_vector_type(2))) float v2f;
typedef __attribute__((ext_vector_type(8))) float v8f;

static constexpr int Nn  = 100000;   // nodes
static constexpr int Ee  = 1600000;  // edges
static constexpr int Bg  = 4096;     // graphs
static constexpr int DIN = 74;       // input feat dim
static constexpr int K1P = 76;       // DIN padded to multiple of 4
static constexpr int H1  = 100;      // gc1 out
static constexpr int DG  = 20;       // gc2 out
static constexpr int D2  = 200;      // 2d descriptor dim
static constexpr int FD  = 4221;     // (DG+1)*(D2+1)
static constexpr int FDP = 4224;     // FD padded to multiple of 4
static constexpr int M1  = 128;
static constexpr int M2  = 32;

// ---------------------------------------------------------------- utilities
__global__ void net2_zero(float* __restrict__ p, int n) {
    int i = blockIdx.x * blockDim.x + threadIdx.x;
    if (i < n) p[i] = 0.0f;
}

// ------------------------------------------------------- edge scatter (COO)
__global__ void net2_degree(const int* __restrict__ dst, float* __restrict__ deg, int ne) {
    int e = blockIdx.x * blockDim.x + threadIdx.x;
    if (e < ne) atomicAdd(&deg[dst[e]], 1.0f);
}

// sum[dst*ldo + d] += x[src*ldi + d]   for d < dim   (one thread per edge*dim)
__global__ void net2_edge_acc(const float* __restrict__ x,
                              const int* __restrict__ src, const int* __restrict__ dst,
                              float* __restrict__ sum, int dim, int ldi, int ldo, int ne) {
    long idx = (long)blockIdx.x * blockDim.x + threadIdx.x;
    if (idx >= (long)ne * dim) return;
    int e = (int)(idx / dim);
    int d = (int)(idx % dim);
    atomicAdd(&sum[(long)dst[e] * ldo + d], x[(long)src[e] * ldi + d]);
}

// mean = deg>0 ? sum/deg : fallback (DGL zero-in-degree semantics), in place
__global__ void net2_mean_fix(float* __restrict__ sum, const float* __restrict__ fallback,
                              const float* __restrict__ deg, int dim, int ldo, int ldi, int nn) {
    long idx = (long)blockIdx.x * blockDim.x + threadIdx.x;
    if (idx >= (long)nn * dim) return;
    int n = (int)(idx / dim);
    int d = (int)(idx % dim);
    float dg = deg[n];
    float v  = (dg > 0.0f) ? sum[(long)n * ldo + d] / dg : fallback[(long)n * ldi + d];
    sum[(long)n * ldo + d] = v;
}

// ---------------------------------------------- weight pre-pack (fragment order)
// Bp holds B[Kpad x Np] zero-padded, reordered so that for K-step kk and lane
// half kbh (0 -> K0/K1, 1 -> K2/K3), column n, the lane's v2f sits at
//   Bp[ (((kk>>2)*2 + kbh) * Np + n) * 2 + {0,1} ]
__global__ void net2_pack_w(const float* __restrict__ Bw, int ldb, int Kreal, int Ncols,
                            float* __restrict__ Bp, int Kpad, int Np) {
    int idx = blockIdx.x * blockDim.x + threadIdx.x;
    if (idx >= Kpad * Np) return;
    int k = idx / Np, n = idx % Np;
    float v = (k < Kreal && n < Ncols) ? Bw[(long)k * ldb + n] : 0.0f;
    long o = ((long)((k >> 2) * 2 + ((k >> 1) & 1)) * Np + n) * 2 + (k & 1);
    Bp[o] = v;
}

// ------------------------------------------------ generic WMMA f32 GEMM tile
// C[M,Ncols] = act(A[M,Kpad] * Bpacked + bias), one wave per 16x16 tile.
// A rows are zero-padded in [Kreal,Kpad); Bp is pre-packed + zero-padded, so
// the inner loop is branch-free and EXEC is trivially all-ones at the WMMA.
__global__ void net2_gemm_wmma(const float* __restrict__ A, int lda, int Kpad,
                               const float* __restrict__ Bp, int Np,
                               const float* __restrict__ bias, int Ncols,
                               float* __restrict__ C, int ldc,
                               int Mtiles, int Ntiles, int do_relu) {
    int wave = threadIdx.x >> 5;
    int lane = threadIdx.x & 31;
    int tile = blockIdx.x * (blockDim.x >> 5) + wave;
    if (tile >= Mtiles * Ntiles) return;          // wave-uniform exit
    int mt = tile / Ntiles, nt = tile % Ntiles;
    int m0 = mt << 4, n0 = nt << 4;
    int l16 = lane & 15;
    int kbh = (lane >> 4) & 1;                    // which K-pair this half-wave holds
    int m   = m0 + l16;                           // A row for this lane
    int n   = n0 + l16;                           // output column for this lane

    // all lda/Np are even -> 8-byte aligned v2f loads
    const v2f* ap = (const v2f*)(A + (long)m * lda) + kbh;
    const v2f* bp = (const v2f*)Bp + (long)kbh * Np + n;

    v8f acc = {};
    int steps = Kpad >> 2;
    for (int s = 0; s < steps; ++s) {
        v2f a = *ap;
        v2f b = *bp;
        acc = __builtin_amdgcn_wmma_f32_16x16x4_f32(false, a, false, b,
                                                    (short)0, acc, false, false);
        ap += 2;        // +4 floats along the A row
        bp += 2 * Np;   // next K-group of the packed B
    }

    bool nok = (n < Ncols);
    float bn = nok ? bias[n] : 0.0f;
    int rbase = m0 + ((lane < 16) ? 0 : 8);       // C/D layout: VGPR r -> row r (+8)
#pragma unroll
    for (int r = 0; r < 8; ++r) {
        float v = acc[r] + bn;
        if (do_relu) v = fmaxf(v, 0.0f);
        if (nok) C[(long)(rbase + r) * ldc + n] = v;
    }
}

// ---------------------------------------------------------- graph readout
__global__ void net2_graph_cnt(const int* __restrict__ gid, float* __restrict__ cnt, int nn) {
    int i = blockIdx.x * blockDim.x + threadIdx.x;
    if (i < nn) atomicAdd(&cnt[gid[i]], 1.0f);
}

__global__ void net2_graph_sum(const float* __restrict__ h, const int* __restrict__ gid,
                               float* __restrict__ sg, int nn) {
    long idx = (long)blockIdx.x * blockDim.x + threadIdx.x;
    if (idx >= (long)nn * DG) return;
    int n = (int)(idx / DG);
    int d = (int)(idx % DG);
    atomicAdd(&sg[(long)gid[n] * DG + d], h[(long)n * DG + d]);
}

__global__ void net2_graph_mean(float* __restrict__ sg, const float* __restrict__ cnt, int nb) {
    int idx = blockIdx.x * blockDim.x + threadIdx.x;
    if (idx >= nb * DG) return;
    int b = idx / DG;
    sg[idx] = sg[idx] / fmaxf(cnt[b], 1.0f);
}

// ---------------------- attention softmax + augmented vectors (LDS reduce)
__global__ void net2_attn(const float* __restrict__ hg, const float* __restrict__ Wattn,
                          const float* __restrict__ desc2d,
                          float* __restrict__ hg_a, float* __restrict__ da_a) {
    __shared__ float shg[DG];
    __shared__ float red[256];
    int b = blockIdx.x, t = threadIdx.x;
    if (t < DG) shg[t] = hg[(long)b * DG + t];
    __syncthreads();
    float logit = 0.0f;
    if (t < D2) {
#pragma unroll
        for (int k = 0; k < DG; ++k) logit += shg[k] * Wattn[k * D2 + t];
    }
    red[t] = (t < D2) ? logit : -3.0e38f;
    __syncthreads();
    for (int s = 128; s > 0; s >>= 1) {
        if (t < s) red[t] = fmaxf(red[t], red[t + s]);
        __syncthreads();
    }
    float mx = red[0];
    __syncthreads();
    float e = (t < D2) ? __expf(logit - mx) : 0.0f;
    red[t] = e;
    __syncthreads();
    for (int s = 128; s > 0; s >>= 1) {
        if (t < s) red[t] += red[t + s];
        __syncthreads();
    }
    float inv = 1.0f / red[0];
    if (t < D2)  da_a[(long)b * (D2 + 1) + t] = e * inv * desc2d[(long)b * D2 + t];
    if (t == D2) da_a[(long)b * (D2 + 1) + D2] = 1.0f;
    if (t < DG)  hg_a[(long)b * (DG + 1) + t] = shg[t];
    if (t == DG) hg_a[(long)b * (DG + 1) + DG] = 1.0f;
}

// fusion[b, i*201+j] = hg_a[b,i] * da_a[b,j], zero-filled to FDP for WMMA K
__global__ void net2_fusion(const float* __restrict__ hg_a, const float* __restrict__ da_a,
                            float* __restrict__ fusion) {
    long idx = (long)blockIdx.x * blockDim.x + threadIdx.x;
    if (idx >= (long)Bg * FDP) return;
    int b = (int)(idx / FDP);
    int j = (int)(idx % FDP);
    float v = 0.0f;
    if (j < FD) {
        int i  = j / (D2 + 1);
        int jj = j % (D2 + 1);
        v = hg_a[(long)b * (DG + 1) + i] * da_a[(long)b * (D2 + 1) + jj];
    }
    fusion[idx] = v;
}

// --------------------------------------------------- BatchNorm (train mode)
__global__ void net2_bn_stats(const float* __restrict__ X, int C, int Rows,
                              float* __restrict__ sum, float* __restrict__ sumsq) {
    __shared__ float s1[256], s2[256];
    int c = blockIdx.x, t = threadIdx.x;
    float a = 0.0f, q = 0.0f;
    for (int r = t; r < Rows; r += 256) {
        float v = X[(long)r * C + c];
        a += v;
        q += v * v;
    }
    s1[t] = a;
    s2[t] = q;
    __syncthreads();
    for (int s = 128; s > 0; s >>= 1) {
        if (t < s) { s1[t] += s1[t + s]; s2[t] += s2[t + s]; }
        __syncthreads();
    }
    if (t == 0) { sum[c] = s1[0]; sumsq[c] = s2[0]; }
}

__global__ void net2_bn_apply(float* __restrict__ X, int C, int Rows,
                              const float* __restrict__ sum, const float* __restrict__ sumsq,
                              const float* __restrict__ gamma, const float* __restrict__ beta) {
    int idx = blockIdx.x * blockDim.x + threadIdx.x;
    if (idx >= Rows * C) return;
    int c = idx % C;
    float inv = 1.0f / (float)Rows;
    float mu  = sum[c] * inv;
    float var = sumsq[c] * inv - mu * mu;
    float v = (X[idx] - mu) * rsqrtf(var + 1e-5f) * gamma[c] + beta[c];
    X[idx] = fmaxf(v, 0.0f);
}

// ------------------------------------------------------------- final layer
__global__ void net2_fc3(const float* __restrict__ A3, const float* __restrict__ W,
                         const float* __restrict__ b, float* __restrict__ out, int nb) {
    int i = blockIdx.x * blockDim.x + threadIdx.x;
    if (i >= nb) return;
    float acc = b[0];
#pragma unroll
    for (int k = 0; k < M2; ++k) acc += A3[(long)i * M2 + k] * W[k];
    out[i] = acc;
}

// ---------------------------------------------------------------------------
extern "C" void kernel_launch(void* const* d_in, const int* in_sizes, int n_in,
                              void* d_out, int out_size, void* d_ws, size_t ws_size,
                              hipStream_t stream) {
    const float* feat   = (const float*)d_in[0];
    const float* desc2d = (const float*)d_in[1];
    /* d_in[2] = desc_3d: unused by the reference */
    const int*   esrc   = (const int*)d_in[3];
    const int*   edst   = (const int*)d_in[4];
    const int*   gids   = (const int*)d_in[5];
    const float* Wgc1   = (const float*)d_in[6];
    const float* bgc1   = (const float*)d_in[7];
    const float* Wgc2   = (const float*)d_in[8];
    const float* bgc2   = (const float*)d_in[9];
    const float* Wattn  = (const float*)d_in[10];
    const float* Wfc1   = (const float*)d_in[11];
    const float* bfc1   = (const float*)d_in[12];
    const float* Wfc2   = (const float*)d_in[13];
    const float* bfc2   = (const float*)d_in[14];
    const float* Wfc3   = (const float*)d_in[15];
    const float* bfc3   = (const float*)d_in[16];
    const float* g1     = (const float*)d_in[17];
    const float* be1    = (const float*)d_in[18];
    const float* g2     = (const float*)d_in[19];
    const float* be2    = (const float*)d_in[20];
    float* out = (float*)d_out;

    // ---- workspace carve-up (floats), 256B-aligned chunks
    float* w = (float*)d_ws;
    size_t off = 0;
    auto alloc = [&](size_t nfloats) {
        float* p = w + off;
        off += (nfloats + 63) & ~(size_t)63;
        return p;
    };
    float* deg   = alloc((size_t)Nn);            // in-degree
    float* agg1  = alloc((size_t)Nn * K1P);      // sum1 -> mean-agg feat [N,76]
    float* h1    = alloc((size_t)Nn * H1);       // gc1 out [N,100]
    float* agg2  = alloc((size_t)Nn * H1);       // sum2 -> mean-agg h1 [N,100]
    float* h2    = alloc((size_t)Nn * DG);       // gc2 out [N,20]
    float* sg    = alloc((size_t)Bg * DG);       // per-graph sum -> mean [B,20]
    float* cnt   = alloc((size_t)Bg);
    float* hg_a  = alloc((size_t)Bg * (DG + 1)); // [B,21]
    float* da_a  = alloc((size_t)Bg * (D2 + 1)); // [B,201]
    float* fus   = alloc((size_t)Bg * FDP);      // [B,4224]
    float* Z1    = alloc((size_t)Bg * M1);       // fc1 out / bn1 in place
    float* Z2    = alloc((size_t)Bg * M2);       // fc2 out / bn2 in place
    float* bn1s  = alloc(M1);
    float* bn1q  = alloc(M1);
    float* bn2s  = alloc(M2);
    float* bn2q  = alloc(M2);
    // packed (zero-padded, fragment-ordered) weights
    const int NP1 = 112, NP2 = 32, NP3 = 128, NP4 = 32;   // Ntiles*16 each
    float* Bp1   = alloc((size_t)K1P * NP1);     // gc1: 76 x 112
    float* Bp2   = alloc((size_t)H1  * NP2);     // gc2: 100 x 32
    float* Bp3   = alloc((size_t)FDP * NP3);     // fc1: 4224 x 128
    float* Bp4   = alloc((size_t)M1  * NP4);     // fc2: 128 x 32

    auto cdiv = [](long a, long b) { return (unsigned)((a + b - 1) / b); };
    const int T = 256;

    // ---- zero the accumulation targets
    net2_zero<<<cdiv(Nn, T), T, 0, stream>>>(deg, Nn);
    net2_zero<<<cdiv((long)Nn * K1P, T), T, 0, stream>>>(agg1, Nn * K1P);
    net2_zero<<<cdiv((long)Nn * H1, T), T, 0, stream>>>(agg2, Nn * H1);
    net2_zero<<<cdiv(Bg * DG, T), T, 0, stream>>>(sg, Bg * DG);
    net2_zero<<<cdiv(Bg, T), T, 0, stream>>>(cnt, Bg);

    // ---- pack all weights into WMMA-fragment order (runs concurrently w/ zeros)
    net2_pack_w<<<cdiv(K1P * NP1, T), T, 0, stream>>>(Wgc1, H1, DIN, H1, Bp1, K1P, NP1);
    net2_pack_w<<<cdiv(H1 * NP2, T), T, 0, stream>>>(Wgc2, DG, H1, DG, Bp2, H1, NP2);
    net2_pack_w<<<cdiv(FDP * NP3, T), T, 0, stream>>>(Wfc1, M1, FD, M1, Bp3, FDP, NP3);
    net2_pack_w<<<cdiv(M1 * NP4, T), T, 0, stream>>>(Wfc2, M2, M1, M2, Bp4, M1, NP4);

    // ---- GCN layer 1: mean agg -> WMMA GEMM [N,76]x[76,112] + relu
    net2_degree<<<cdiv(Ee, T), T, 0, stream>>>(edst, deg, Ee);
    net2_edge_acc<<<cdiv((long)Ee * DIN, T), T, 0, stream>>>(feat, esrc, edst, agg1,
                                                            DIN, DIN, K1P, Ee);
    net2_mean_fix<<<cdiv((long)Nn * DIN, T), T, 0, stream>>>(agg1, feat, deg,
                                                            DIN, K1P, DIN, Nn);
    {
        int Mt = Nn / 16, Nt = NP1 / 16;         // 6250 x 7 tiles
        net2_gemm_wmma<<<cdiv((long)Mt * Nt, 8), T, 0, stream>>>(
            agg1, K1P, K1P, Bp1, NP1, bgc1, H1, h1, H1, Mt, Nt, 1);
    }

    // ---- GCN layer 2: mean agg -> WMMA GEMM [N,100]x[100,32] + relu
    net2_edge_acc<<<cdiv((long)Ee * H1, T), T, 0, stream>>>(h1, esrc, edst, agg2,
                                                           H1, H1, H1, Ee);
    net2_mean_fix<<<cdiv((long)Nn * H1, T), T, 0, stream>>>(agg2, h1, deg,
                                                           H1, H1, H1, Nn);
    {
        int Mt = Nn / 16, Nt = NP2 / 16;         // 6250 x 2 tiles
        net2_gemm_wmma<<<cdiv((long)Mt * Nt, 8), T, 0, stream>>>(
            agg2, H1, H1, Bp2, NP2, bgc2, DG, h2, DG, Mt, Nt, 1);
    }

    // ---- per-graph mean readout
    net2_graph_cnt<<<cdiv(Nn, T), T, 0, stream>>>(gids, cnt, Nn);
    net2_graph_sum<<<cdiv((long)Nn * DG, T), T, 0, stream>>>(h2, gids, sg, Nn);
    net2_graph_mean<<<cdiv(Bg * DG, T), T, 0, stream>>>(sg, cnt, Bg);

    // ---- attention softmax + augmented vectors + fusion outer product
    net2_attn<<<Bg, T, 0, stream>>>(sg, Wattn, desc2d, hg_a, da_a);
    net2_fusion<<<cdiv((long)Bg * FDP, T), T, 0, stream>>>(hg_a, da_a, fus);

    // ---- MLP: fc1 (WMMA, K=4224) -> BN -> relu
    {
        int Mt = Bg / 16, Nt = NP3 / 16;         // 256 x 8 tiles
        net2_gemm_wmma<<<cdiv((long)Mt * Nt, 8), T, 0, stream>>>(
            fus, FDP, FDP, Bp3, NP3, bfc1, M1, Z1, M1, Mt, Nt, 0);
    }
    net2_bn_stats<<<M1, T, 0, stream>>>(Z1, M1, Bg, bn1s, bn1q);
    net2_bn_apply<<<cdiv(Bg * M1, T), T, 0, stream>>>(Z1, M1, Bg, bn1s, bn1q, g1, be1);

    // ---- fc2 (WMMA, K=128) -> BN -> relu
    {
        int Mt = Bg / 16, Nt = NP4 / 16;         // 256 x 2 tiles
        net2_gemm_wmma<<<cdiv((long)Mt * Nt, 8), T, 0, stream>>>(
            Z1, M1, M1, Bp4, NP4, bfc2, M2, Z2, M2, Mt, Nt, 0);
    }
    net2_bn_stats<<<M2, T, 0, stream>>>(Z2, M2, Bg, bn2s, bn2q);
    net2_bn_apply<<<cdiv(Bg * M2, T), T, 0, stream>>>(Z2, M2, Bg, bn2s, bn2q, g2, be2);

    // ---- fc3
    net2_fc3<<<cdiv(Bg, T), T, 0, stream>>>(Z2, Wfc3, bfc3, out, Bg);
}